// InverseTransform_5446018531803
// MI455X (gfx1250) — compile-verified
//
#include <hip/hip_runtime.h>

typedef float v2f __attribute__((ext_vector_type(2)));
typedef float v8f __attribute__((ext_vector_type(8)));

#define B_   16
#define F_   2048
#define NBIN 257
#define WIN  1024
#define BLK  256
#define KJ   4   // WIN / BLK
#define ZROW 512 // floats of zero scratch in d_ws

// Tiny graph-safe init kernel: zero the pad row used for out-of-range A rows.
__global__ void itrans_zero_ws(float* __restrict__ z) {
    z[threadIdx.x] = 0.0f;   // one block of ZROW threads
}

// One wave computes a 16(t) x 32(c) tile of out[b, t, c] via
//   out[b,t,c] = sum_j  x[b, t-j, :] . mod[:, j*256 + c]   (complex-real GEMM, fused fold)
// using V_WMMA_F32_16X16X4_F32. Each A operand feeds two adjacent 16-wide B
// column tiles (3 load-dwords per WMMA instead of 4); four independent
// accumulators (real/imag x 2 tiles) keep the XDL pipe busy. Out-of-range A
// rows read a zeroed scratch row so the hot loop has zero branches.
// Block = 256 threads = 8 waves x 32 cols = 256 cols; grid = (F/16, 1, B).
__global__ __launch_bounds__(256)
void itrans_fold_wmma(const float* __restrict__ xr,
                      const float* __restrict__ xi,
                      const float* __restrict__ mr,
                      const float* __restrict__ mi,
                      const float* __restrict__ st,
                      const float* __restrict__ zrow,
                      float* __restrict__ out)
{
    const int lane = threadIdx.x & 31;
    const int wave = threadIdx.x >> 5;
    const int t0   = blockIdx.x * 16;   // output frame tile base
    const int c0   = wave * 32;         // output column tile base (two 16-wide tiles)
    const int b    = blockIdx.z;

    const int mrow = lane & 15;   // A row (M) index for this lane
    const int hi   = lane >> 4;   // lanes 16-31 carry the K+2 / K+3 halves
    const int n    = lane & 15;   // B column (N) index for this lane

    v8f accR0 = {}, accI0 = {};
    v8f accR1 = {}, accI1 = {};

    for (int j = 0; j < KJ; ++j) {
        const int frame = t0 + mrow - j;              // A row = x frame (t - j)
        const bool rok  = (frame >= 0);               // frame < F_ always holds
        const size_t rbase = ((size_t)b * F_ + (rok ? frame : 0)) * NBIN;

        // Per-lane base pointers; invalid rows point at the zero row so all
        // loads in the hot loop are unconditional.
        const float* ar_p = (rok ? xr + rbase : zrow) + 2 * hi;
        const float* ai_p = (rok ? xi + rbase : zrow) + 2 * hi;
        const int w = j * BLK + c0 + n;               // mod column, first tile
        const float* br_p = mr + w + (size_t)(2 * hi) * WIN;
        const float* bi_p = mi + w + (size_t)(2 * hi) * WIN;

        // Main K loop: k = 0..255, 4 per WMMA, unrolled x4.
        #pragma unroll 4
        for (int k0 = 0; k0 < 256; k0 += 4) {
            v2f ar, ai, br0, bi0, br1, bi1;
            ar.x  = ar_p[k0];
            ar.y  = ar_p[k0 + 1];
            ai.x  = ai_p[k0];
            ai.y  = ai_p[k0 + 1];
            br0.x = br_p[(size_t)k0 * WIN];
            br0.y = br_p[(size_t)(k0 + 1) * WIN];
            bi0.x = bi_p[(size_t)k0 * WIN];
            bi0.y = bi_p[(size_t)(k0 + 1) * WIN];
            br1.x = br_p[(size_t)k0 * WIN + 16];        // second tile: +64B imm
            br1.y = br_p[(size_t)(k0 + 1) * WIN + 16];
            bi1.x = bi_p[(size_t)k0 * WIN + 16];
            bi1.y = bi_p[(size_t)(k0 + 1) * WIN + 16];
            accR0 = __builtin_amdgcn_wmma_f32_16x16x4_f32(
                        false, ar, false, br0, (short)0, accR0, false, false);
            accI0 = __builtin_amdgcn_wmma_f32_16x16x4_f32(
                        false, ai, false, bi0, (short)0, accI0, false, false);
            accR1 = __builtin_amdgcn_wmma_f32_16x16x4_f32(
                        false, ar, false, br1, (short)0, accR1, false, false);
            accI1 = __builtin_amdgcn_wmma_f32_16x16x4_f32(
                        false, ai, false, bi1, (short)0, accI1, false, false);
        }

        // Tail: k = 256 (K=257). Only lanes hi=0, component .x carry data;
        // everything else reads the zero row / constant zero.
        {
            const bool tok = rok && (hi == 0);
            const float* tar = tok ? (xr + rbase + 256) : zrow;
            const float* tai = tok ? (xi + rbase + 256) : zrow;
            const float* tbr = (hi == 0) ? (mr + (size_t)256 * WIN + w) : zrow;
            const float* tbi = (hi == 0) ? (mi + (size_t)256 * WIN + w) : zrow;
            v2f ar, ai, br0, bi0, br1, bi1;
            ar.x  = tar[0];  ar.y  = 0.0f;
            ai.x  = tai[0];  ai.y  = 0.0f;
            br0.x = tbr[0];  br0.y = 0.0f;
            bi0.x = tbi[0];  bi0.y = 0.0f;
            br1.x = tbr[16]; br1.y = 0.0f;   // zrow[16] is also zero
            bi1.x = tbi[16]; bi1.y = 0.0f;
            accR0 = __builtin_amdgcn_wmma_f32_16x16x4_f32(
                        false, ar, false, br0, (short)0, accR0, false, false);
            accI0 = __builtin_amdgcn_wmma_f32_16x16x4_f32(
                        false, ai, false, bi0, (short)0, accI0, false, false);
            accR1 = __builtin_amdgcn_wmma_f32_16x16x4_f32(
                        false, ar, false, br1, (short)0, accR1, false, false);
            accI1 = __builtin_amdgcn_wmma_f32_16x16x4_f32(
                        false, ai, false, bi1, (short)0, accI1, false, false);
        }
    }

    // Epilogue: C layout is VGPR r, lane L -> M = r + 8*hi, N = L&15.
    // out = accR - accI for both column tiles, plus the streamed-state add
    // (state lands on output frames 0..2 only).
    #pragma unroll
    for (int r = 0; r < 8; ++r) {
        const int t  = t0 + r + 8 * hi;
        const int c  = c0 + n;
        float v0 = accR0[r] - accI0[r];
        float v1 = accR1[r] - accI1[r];
        if (t < KJ - 1) {
            const float* srow = st + ((size_t)b * (KJ - 1) + t) * BLK;
            v0 += srow[c];
            v1 += srow[c + 16];
        }
        float* orow = out + ((size_t)b * F_ + t) * BLK;
        orow[c]      = v0;
        orow[c + 16] = v1;
    }
}

extern "C" void kernel_launch(void* const* d_in, const int* in_sizes, int n_in,
                              void* d_out, int out_size, void* d_ws, size_t ws_size,
                              hipStream_t stream) {
    (void)in_sizes; (void)n_in; (void)out_size; (void)ws_size;
    const float* xr = (const float*)d_in[0];   // [B, F, NBIN]
    const float* xi = (const float*)d_in[1];   // [B, F, NBIN]
    const float* mr = (const float*)d_in[2];   // [NBIN, WIN]
    const float* mi = (const float*)d_in[3];   // [NBIN, WIN]
    const float* st = (const float*)d_in[4];   // [B, 3, BLK]
    // d_in[5] = block_size scalar (compile-time BLK here)
    float* out = (float*)d_out;                // [B, F, BLK]
    float* zrow = (float*)d_ws;                // ZROW zeroed floats

    hipLaunchKernelGGL(itrans_zero_ws, dim3(1), dim3(ZROW), 0, stream, zrow);

    dim3 grid(F_ / 16, 1, B_);
    dim3 block(256);
    hipLaunchKernelGGL(itrans_fold_wmma, grid, block, 0, stream,
                       xr, xi, mr, mi, st, zrow, out);
}